// NetGIN_9234179686416
// MI455X (gfx1250) — compile-verified
//
#include <hip/hip_runtime.h>

#define N_NODES  500000
#define N_EDGES  16000000
#define N_GRAPHS 5000
#define TPB      8   // tiles (waves) per block in the MLP kernel

typedef __attribute__((ext_vector_type(2))) float v2f;
typedef __attribute__((ext_vector_type(8))) float v8f;

// ---------------------------------------------------------------------------
// utility kernels
// ---------------------------------------------------------------------------
__global__ void zero_kernel(float* __restrict__ p, int n) {
  int i = blockIdx.x * blockDim.x + threadIdx.x;
  if (i < n) p[i] = 0.0f;
}

__global__ void copy4_kernel(const float4* __restrict__ s, float4* __restrict__ d, int n4) {
  int i = blockIdx.x * blockDim.x + threadIdx.x;
  if (i < n4) d[i] = s[i];
}

__global__ void count_kernel(const int* __restrict__ ng, float* __restrict__ counts, int n) {
  int i = blockIdx.x * blockDim.x + threadIdx.x;
  if (i < n) unsafeAtomicAdd(&counts[ng[i]], 1.0f);
}

// ---------------------------------------------------------------------------
// layer-1 scatter: y64[dst] += h[src]  (64 floats/edge, 4 threads/edge)
// ---------------------------------------------------------------------------
__global__ void scatter64_kernel(const int* __restrict__ src, const int* __restrict__ dst,
                                 const float* __restrict__ h, float* __restrict__ y) {
  int t = blockIdx.x * blockDim.x + threadIdx.x;   // 4*E threads
  int e = t >> 2;
  if (e >= N_EDGES) return;
  int c = (t & 3) << 4;                            // 0,16,32,48
  const float4* hp = reinterpret_cast<const float4*>(h + (size_t)src[e] * 64 + c);
  float*        yp = y + (size_t)dst[e] * 64 + c;
  float4 v0 = hp[0], v1 = hp[1], v2 = hp[2], v3 = hp[3];
  unsafeAtomicAdd(yp + 0,  v0.x); unsafeAtomicAdd(yp + 1,  v0.y);
  unsafeAtomicAdd(yp + 2,  v0.z); unsafeAtomicAdd(yp + 3,  v0.w);
  unsafeAtomicAdd(yp + 4,  v1.x); unsafeAtomicAdd(yp + 5,  v1.y);
  unsafeAtomicAdd(yp + 6,  v1.z); unsafeAtomicAdd(yp + 7,  v1.w);
  unsafeAtomicAdd(yp + 8,  v2.x); unsafeAtomicAdd(yp + 9,  v2.y);
  unsafeAtomicAdd(yp + 10, v2.z); unsafeAtomicAdd(yp + 11, v2.w);
  unsafeAtomicAdd(yp + 12, v3.x); unsafeAtomicAdd(yp + 13, v3.y);
  unsafeAtomicAdd(yp + 14, v3.z); unsafeAtomicAdd(yp + 15, v3.w);
}

// ---------------------------------------------------------------------------
// layers 2-5 scatter: y16[dst, 0:10] += x[src, 0:10]   (stride-16 storage)
// ---------------------------------------------------------------------------
__global__ void scatter16_kernel(const int* __restrict__ src, const int* __restrict__ dst,
                                 const float* __restrict__ x, float* __restrict__ y) {
  int e = blockIdx.x * blockDim.x + threadIdx.x;
  if (e >= N_EDGES) return;
  const float* xp = x + (size_t)src[e] * 16;
  float*       yp = y + (size_t)dst[e] * 16;
  float4 v0 = *reinterpret_cast<const float4*>(xp + 0);
  float4 v1 = *reinterpret_cast<const float4*>(xp + 4);
  float2 v2 = *reinterpret_cast<const float2*>(xp + 8);
  unsafeAtomicAdd(yp + 0, v0.x); unsafeAtomicAdd(yp + 1, v0.y);
  unsafeAtomicAdd(yp + 2, v0.z); unsafeAtomicAdd(yp + 3, v0.w);
  unsafeAtomicAdd(yp + 4, v1.x); unsafeAtomicAdd(yp + 5, v1.y);
  unsafeAtomicAdd(yp + 6, v1.z); unsafeAtomicAdd(yp + 7, v1.w);
  unsafeAtomicAdd(yp + 8, v2.x); unsafeAtomicAdd(yp + 9, v2.y);
}

// ---------------------------------------------------------------------------
// MLP via V_WMMA_F32_16X16X4_F32: one wave = one 16-node tile, 8 waves/block.
//   xout[tile, 0:16] = relu( relu(Y @ W1p + b1p) @ W2p + b2p )
// Padded weight tables are built once per block in LDS (as float2 K-pairs),
// so the unrolled K-loops are pure load_b64 + ds_load_b64 + v_wmma with EXEC
// all-ones (no divergent weight predication in the hot path).
// ---------------------------------------------------------------------------
template <int LDY, int DIN>
__global__ __launch_bounds__(32 * TPB) void mlp_wmma_kernel(
    const float* __restrict__ y,
    const float* __restrict__ W1, const float* __restrict__ b1,
    const float* __restrict__ W2, const float* __restrict__ b2,
    float* __restrict__ xout, int ntiles)
{
  __shared__ float2 wb1[LDY / 2][16];    // wb1[p][n] = {W1p[2p][n], W1p[2p+1][n]}
  __shared__ float2 wb2[8][16];          // wb2[p][n] = {W2p[2p][n], W2p[2p+1][n]}
  __shared__ float  bias[2][16];         // padded b1, b2
  __shared__ float  hb[TPB][16][18];     // per-wave D->A bounce (row stride 18: float2-aligned)

  const int tid  = threadIdx.x;
  const int lane = tid & 31;
  const int w    = tid >> 5;

  // ---- cooperative padded-weight build (branches OK here, before any WMMA) ----
  for (int idx = tid; idx < (LDY / 2) * 16; idx += 32 * TPB) {
    int p = idx >> 4, n = idx & 15;
    float x0 = (n < 10 && 2 * p     < DIN) ? W1[(2 * p)     * 10 + n] : 0.0f;
    float x1 = (n < 10 && 2 * p + 1 < DIN) ? W1[(2 * p + 1) * 10 + n] : 0.0f;
    wb1[p][n] = make_float2(x0, x1);
  }
  for (int idx = tid; idx < 8 * 16; idx += 32 * TPB) {
    int p = idx >> 4, n = idx & 15;
    float x0 = (n < 10 && 2 * p     < 10) ? W2[(2 * p)     * 10 + n] : 0.0f;
    float x1 = (n < 10 && 2 * p + 1 < 10) ? W2[(2 * p + 1) * 10 + n] : 0.0f;
    wb2[p][n] = make_float2(x0, x1);
  }
  if (tid < 32) {
    int n = tid & 15, which = tid >> 4;
    bias[which][n] = (n < 10) ? (which ? b2[n] : b1[n]) : 0.0f;
  }
  __syncthreads();

  // tail waves clamp to the last tile (recompute duplicates; identical stores)
  int tile = blockIdx.x * TPB + w;
  if (tile >= ntiles) tile = ntiles - 1;
  const int m   = lane & 15;   // A row / B-C-D column
  const int hi  = lane >> 4;
  const int n   = m;
  const int row0 = tile * 16;

  // ---- GEMM1: C(16x16) = Y(16 x LDY) @ W1p(LDY x 16) ----
  v8f c = {};
  const float* yrow = y + (size_t)(row0 + m) * LDY;
#pragma unroll
  for (int k = 0; k < LDY; k += 4) {
    float2 ap = *reinterpret_cast<const float2*>(yrow + k + 2 * hi);
    float2 bp = wb1[(k >> 1) + hi][n];
    v2f a; a.x = ap.x; a.y = ap.y;
    v2f b; b.x = bp.x; b.y = bp.y;
    c = __builtin_amdgcn_wmma_f32_16x16x4_f32(false, a, false, b, (short)0, c,
                                              false, false);
  }
  const float bias1 = bias[0][n];
#pragma unroll
  for (int v = 0; v < 8; ++v) {
    float t = c[v] + bias1;
    hb[w][v + 8 * hi][n] = t > 0.0f ? t : 0.0f;   // D layout: lane=col, VGPR=row
  }
  // wave-local LDS RAW ordering (same-wave DS ops are in-order; drain DScnt)
  asm volatile("s_wait_dscnt 0x0" ::: "memory");

  // ---- GEMM2: D(16x16) = Hmid(16x16) @ W2p(16x16) ----
  v8f d = {};
#pragma unroll
  for (int k = 0; k < 16; k += 4) {
    float2 ap = *reinterpret_cast<const float2*>(&hb[w][m][k + 2 * hi]);
    float2 bp = wb2[(k >> 1) + hi][n];
    v2f a; a.x = ap.x; a.y = ap.y;
    v2f b; b.x = bp.x; b.y = bp.y;
    d = __builtin_amdgcn_wmma_f32_16x16x4_f32(false, a, false, b, (short)0, d,
                                              false, false);
  }
  const float bias2 = bias[1][n];
#pragma unroll
  for (int v = 0; v < 8; ++v) {
    float t = d[v] + bias2;
    t = t > 0.0f ? t : 0.0f;                         // outer ReLU
    xout[(size_t)(row0 + v + 8 * hi) * 16 + n] = t;  // cols 10..15 get 0
  }
}

// ---------------------------------------------------------------------------
// per-layer readout: sacc[g] += dot(x[n,0:10], l) / max(count[g],1)
// ---------------------------------------------------------------------------
__global__ void readout_kernel(const float* __restrict__ x, const float* __restrict__ l,
                               const int* __restrict__ ng, const float* __restrict__ counts,
                               float* __restrict__ sacc, int nn) {
  int i = blockIdx.x * blockDim.x + threadIdx.x;
  if (i >= nn) return;
  const float* xr = x + (size_t)i * 16;
  float dot = 0.0f;
#pragma unroll
  for (int j = 0; j < 10; ++j) dot += xr[j] * l[j];
  int g = ng[i];
  float cnt = counts[g];
  cnt = cnt > 1.0f ? cnt : 1.0f;
  unsafeAtomicAdd(&sacc[g], dot / cnt);
}

__global__ void sigmoid_kernel(const float* __restrict__ sacc, float* __restrict__ out, int g) {
  int i = blockIdx.x * blockDim.x + threadIdx.x;
  if (i < g) out[i] = 1.0f / (1.0f + __expf(-sacc[i]));
}

// ---------------------------------------------------------------------------
// launch
// ---------------------------------------------------------------------------
extern "C" void kernel_launch(void* const* d_in, const int* in_sizes, int n_in,
                              void* d_out, int out_size, void* d_ws, size_t ws_size,
                              hipStream_t stream) {
  const float* h   = (const float*)d_in[0];
  const int*   src = (const int*)d_in[1];
  const int*   dst = (const int*)d_in[2];
  const int*   ng  = (const int*)d_in[3];
  const float *W1[5], *b1[5], *W2[5], *b2[5], *lv[5];
  for (int i = 0; i < 5; ++i) {
    W1[i] = (const float*)d_in[4 + 5 * i];
    b1[i] = (const float*)d_in[5 + 5 * i];
    W2[i] = (const float*)d_in[6 + 5 * i];
    b2[i] = (const float*)d_in[7 + 5 * i];
    lv[i] = (const float*)d_in[8 + 5 * i];
  }
  float* out = (float*)d_out;

  // workspace layout (floats)
  float* ws     = (float*)d_ws;
  float* y64    = ws;                                  // N*64
  float* xA     = y64 + (size_t)N_NODES * 64;          // N*16
  float* xB     = xA  + (size_t)N_NODES * 16;          // N*16
  float* y16    = xB  + (size_t)N_NODES * 16;          // N*16
  float* counts = y16 + (size_t)N_NODES * 16;          // G
  float* sacc   = counts + N_GRAPHS;                   // G

  const int B = 256;
  const int ntiles = N_NODES / 16;            // 31250 (N divisible by 16)
  const int mlp_blocks = (ntiles + TPB - 1) / TPB;

  // zero counts + sacc (contiguous)
  zero_kernel<<<(2 * N_GRAPHS + B - 1) / B, B, 0, stream>>>(counts, 2 * N_GRAPHS);
  // graph sizes
  count_kernel<<<(N_NODES + B - 1) / B, B, 0, stream>>>(ng, counts, N_NODES);

  // ---- layer 1 (din = 64) ----
  {
    int n4 = N_NODES * 64 / 4;
    copy4_kernel<<<(n4 + B - 1) / B, B, 0, stream>>>((const float4*)h, (float4*)y64, n4);
    long long t = (long long)N_EDGES * 4;
    scatter64_kernel<<<(unsigned)((t + B - 1) / B), B, 0, stream>>>(src, dst, h, y64);
    mlp_wmma_kernel<64, 64><<<mlp_blocks, 32 * TPB, 0, stream>>>(
        y64, W1[0], b1[0], W2[0], b2[0], xA, ntiles);
    readout_kernel<<<(N_NODES + B - 1) / B, B, 0, stream>>>(xA, lv[0], ng, counts,
                                                            sacc, N_NODES);
  }

  // ---- layers 2..5 (din = 10, stride-16 storage) ----
  float* xcur = xA;
  float* xnxt = xB;
  for (int i = 1; i < 5; ++i) {
    int n4 = N_NODES * 16 / 4;
    copy4_kernel<<<(n4 + B - 1) / B, B, 0, stream>>>((const float4*)xcur, (float4*)y16, n4);
    scatter16_kernel<<<(N_EDGES + B - 1) / B, B, 0, stream>>>(src, dst, xcur, y16);
    mlp_wmma_kernel<16, 10><<<mlp_blocks, 32 * TPB, 0, stream>>>(
        y16, W1[i], b1[i], W2[i], b2[i], xnxt, ntiles);
    readout_kernel<<<(N_NODES + B - 1) / B, B, 0, stream>>>(xnxt, lv[i], ng, counts,
                                                            sacc, N_NODES);
    float* tmp = xcur; xcur = xnxt; xnxt = tmp;
  }

  sigmoid_kernel<<<(N_GRAPHS + B - 1) / B, B, 0, stream>>>(sacc, out, N_GRAPHS);
}